// GCONV_60224031425327
// MI455X (gfx1250) — compile-verified
//
#include <hip/hip_runtime.h>

#define HID 64
#define NC 3
#define WSTRIDE 68   // padded LDS row stride (floats): conflict-free b64 reads

typedef __attribute__((ext_vector_type(2))) float v2f;
typedef __attribute__((ext_vector_type(8))) float v8f;

// Non-returning f32 atomic add straight to the global path (L2 atomic unit).
// Avoids any CAS-loop lowering; tracked with STOREcnt, s_endpgm waits for it.
__device__ __forceinline__ void gadd_f32(float* p, float v) {
    asm volatile("global_atomic_add_f32 %0, %1, off" :: "v"(p), "v"(v) : "memory");
}

__global__ void fill_zero(float* __restrict__ p, int n) {
    int i = blockIdx.x * blockDim.x + threadIdx.x;
    if (i < n) p[i] = 0.0f;
}

// One wave per edge: lane L handles channels {2L, 2L+1}. The 256 B source row
// is a contiguous coalesced load; destination adds are L2-resident atomics.
__global__ __launch_bounds__(256) void scatter_edges(
        const float* __restrict__ h, const long long* __restrict__ src,
        const long long* __restrict__ dst, float* __restrict__ agg, int nE) {
    int wid  = (blockIdx.x * blockDim.x + threadIdx.x) >> 5;
    int lane = threadIdx.x & 31;
    if (wid >= nE) return;
    long long s = src[wid];
    long long d = dst[wid];
    const float2 hv = *(const float2*)(h + s * HID + lane * 2);
    float* p = agg + d * HID + lane * 2;
    gadd_f32(p,     hv.x);
    gadd_f32(p + 1, hv.y);
}

// Fused GraphConv update: hout = relu(agg @ Wrel + hin @ Wroot + bias)
// One wave32 per 16-row node tile, full K=64 via 16 chained
// V_WMMA_F32_16X16X4_F32 steps per matrix, 4 column tiles of 16.
// Weights are staged transposed in LDS (sW[n*WSTRIDE + k]) so B fragments are
// single ds_load_b64 per matrix per k-step, bank-conflict free.
__global__ __launch_bounds__(256) void gconv_gemm(
        const float* __restrict__ agg, const float* hin,
        const float* __restrict__ Wrel, const float* __restrict__ Wroot,
        const float* __restrict__ bias, float* hout, int nTiles) {
    __shared__ float sRel[HID * WSTRIDE];
    __shared__ float sRoot[HID * WSTRIDE];
    __shared__ float sB[HID];

    for (int i = threadIdx.x; i < HID * HID; i += blockDim.x) {
        int k = i >> 6, n = i & 63;           // W is [k][n] row-major
        sRel [n * WSTRIDE + k] = Wrel[i];     // store transposed
        sRoot[n * WSTRIDE + k] = Wroot[i];
    }
    if (threadIdx.x < HID) sB[threadIdx.x] = bias[threadIdx.x];
    __syncthreads();

    int wave = threadIdx.x >> 5;
    int lane = threadIdx.x & 31;
    int tile = blockIdx.x * 8 + wave;
    if (tile >= nTiles) return;               // wave-uniform: EXEC stays all-1s

    int row = lane & 15;                      // A: M index / B,C: N index
    int kh  = lane >> 4;                      // 0 -> K{0,1}, 1 -> K{2,3}
    long long base = (long long)tile * 16;

    // Cache both A operands' fragments for all 16 k-blocks (reused 4x).
    v2f aA[16], aH[16];
    const float* pA = agg + (base + row) * HID + 2 * kh;
    const float* pH = hin + (base + row) * HID + 2 * kh;
#pragma unroll
    for (int kb = 0; kb < 16; ++kb) {
        aA[kb] = *(const v2f*)(pA + kb * 4);
        aH[kb] = *(const v2f*)(pH + kb * 4);
    }

#pragma unroll
    for (int ct = 0; ct < 4; ++ct) {
        int col = ct * 16 + row;
        v8f c = {};
#pragma unroll
        for (int kb = 0; kb < 16; ++kb) {
            int k0 = kb * 4 + 2 * kh;
            v2f bRel  = *(const v2f*)(&sRel [col * WSTRIDE + k0]);
            v2f bRoot = *(const v2f*)(&sRoot[col * WSTRIDE + k0]);
            c = __builtin_amdgcn_wmma_f32_16x16x4_f32(false, aA[kb], false, bRel,
                                                      (short)0, c, false, false);
            c = __builtin_amdgcn_wmma_f32_16x16x4_f32(false, aH[kb], false, bRoot,
                                                      (short)0, c, false, false);
        }
        float bb = sB[col];
        // C/D layout: lanes 0-15 hold M=j, lanes 16-31 hold M=8+j, N=lane&15
        float* outp = hout + (base + 8 * kh) * HID + col;
#pragma unroll
        for (int j = 0; j < 8; ++j) {
            float v = c[j] + bb;
            outp[j * HID] = v > 0.0f ? v : 0.0f;
        }
    }
}

// One wave per node: sum-pool node features into pooled[batch[n]].
__global__ __launch_bounds__(256) void pool_nodes(
        const float* __restrict__ h, const long long* __restrict__ batch,
        float* __restrict__ pooled, int nNodes) {
    int wid  = (blockIdx.x * blockDim.x + threadIdx.x) >> 5;
    int lane = threadIdx.x & 31;
    if (wid >= nNodes) return;
    long long g = batch[wid];
    const float2 hv = *(const float2*)(h + (long long)wid * HID + lane * 2);
    float* p = pooled + g * HID + lane * 2;
    gadd_f32(p,     hv.x);
    gadd_f32(p + 1, hv.y);
}

// One 128-thread block per graph: hidden = relu(pooled@W1+b1), out = sigmoid(hidden@W2+b2)
__global__ __launch_bounds__(128) void mlp_readout(
        const float* __restrict__ pooled, const float* __restrict__ W1,
        const float* __restrict__ b1, const float* __restrict__ W2,
        const float* __restrict__ b2, float* __restrict__ out) {
    __shared__ float sP[HID];
    __shared__ float sH[2 * HID];
    int g = blockIdx.x;
    if (threadIdx.x < HID) sP[threadIdx.x] = pooled[g * HID + threadIdx.x];
    __syncthreads();
    int j = threadIdx.x;                       // 0..127 hidden units
    float acc = b1[j];
#pragma unroll 8
    for (int k = 0; k < HID; ++k) acc += sP[k] * W1[k * 2 * HID + j];
    sH[j] = acc > 0.0f ? acc : 0.0f;
    __syncthreads();
    if (threadIdx.x < 2) {
        float a = b2[threadIdx.x];
        for (int k = 0; k < 2 * HID; ++k) a += sH[k] * W2[k * 2 + threadIdx.x];
        out[g * 2 + threadIdx.x] = 1.0f / (1.0f + __expf(-a));
    }
}

extern "C" void kernel_launch(void* const* d_in, const int* in_sizes, int n_in,
                              void* d_out, int out_size, void* d_ws, size_t ws_size,
                              hipStream_t stream) {
    const float*     x     = (const float*)d_in[0];
    const long long* ei    = (const long long*)d_in[1];   // [2, E] int64
    const long long* batch = (const long long*)d_in[2];   // [N]    int64
    const float*     Wrel  = (const float*)d_in[3];       // [3,64,64]
    const float*     Wroot = (const float*)d_in[4];       // [3,64,64]
    const float*     bconv = (const float*)d_in[5];       // [3,64]
    const float*     W1    = (const float*)d_in[6];       // [64,128]
    const float*     b1    = (const float*)d_in[7];
    const float*     W2    = (const float*)d_in[8];       // [128,2]
    const float*     b2    = (const float*)d_in[9];

    const int nNodes  = in_sizes[0] / HID;       // 100000
    const int nEdges  = in_sizes[1] / 2;         // 1600000
    const int nGraphs = out_size / (2 + HID);    // 512

    float* out    = (float*)d_out;                       // [G,2]
    float* pooled = out + (size_t)nGraphs * 2;           // [G,64] (second output)

    // Workspace: agg buffer + h ping buffer (2 x 25.6 MB; both L2-resident)
    float* agg = (float*)d_ws;
    float* h   = agg + (size_t)nNodes * HID;

    const int feat   = nNodes * HID;
    const int nTiles = nNodes / 16;              // 100000 % 16 == 0
    const long long* src = ei;
    const long long* dst = ei + nEdges;

    const float* hin = x;
    for (int l = 0; l < NC; ++l) {
        fill_zero    <<<(feat + 255) / 256, 256, 0, stream>>>(agg, feat);
        scatter_edges<<<(nEdges + 7) / 8,   256, 0, stream>>>(hin, src, dst, agg, nEdges);
        gconv_gemm   <<<(nTiles + 7) / 8,   256, 0, stream>>>(
            agg, hin, Wrel + l * HID * HID, Wroot + l * HID * HID,
            bconv + l * HID, h, nTiles);
        hin = h;   // layers 1,2 update h in place (safe: per-wave read-then-write)
    }

    fill_zero <<<(nGraphs * HID + 255) / 256, 256, 0, stream>>>(pooled, nGraphs * HID);
    pool_nodes<<<(nNodes + 7) / 8,            256, 0, stream>>>(h, batch, pooled, nNodes);
    mlp_readout<<<nGraphs, 128, 0, stream>>>(pooled, W1, b1, W2, b2, out);
}